// SPIRED_Model_47029891891568
// MI455X (gfx1250) — compile-verified
//
#include <hip/hip_runtime.h>

typedef float v2f __attribute__((ext_vector_type(2)));
typedef float v8f __attribute__((ext_vector_type(8)));
typedef float f4  __attribute__((ext_vector_type(4)));

#define L     320
#define LL    (L * L)          // 102400
#define DDIM  128
#define BINS  32
#define EPS   1e-8f

// ---------------------------------------------------------------------------
// Kernel 1: z = pair_feats + emb_table[clip(residx[j]-residx[i])+33 or 0]
// Bandwidth-bound: b128 loads/stores, one float4 per thread.
// pair_feats / z are streamed once -> non-temporal hints; emb_table is hot.
// total float4 = 320*320*128/4 = 3,276,800 -> 12800 blocks x 256 threads
// ---------------------------------------------------------------------------
__global__ void __launch_bounds__(256) z_kernel(const int* __restrict__ residx,
                                                const unsigned char* __restrict__ mask,
                                                const float* __restrict__ emb,
                                                const float* __restrict__ pf,
                                                float* __restrict__ out)
{
    int t  = blockIdx.x * blockDim.x + threadIdx.x;   // float4 index
    int p  = t >> 5;                                   // pair index (i*L + j)
    int d4 = t & 31;
    int i  = p / L;
    int j  = p - i * L;

    int diff = residx[j] - residx[i];
    diff = min(max(diff, -BINS), BINS) + (BINS + 1);
    if (!(mask[i] && mask[j])) diff = 0;

    const f4* pf4 = (const f4*)pf;
    const f4* e4  = (const f4*)emb;
    f4 v = __builtin_nontemporal_load(pf4 + t);        // streamed, no reuse
    f4 e = e4[diff * (DDIM / 4) + d4];                 // hot 33KB table: cached
    v.x += e.x; v.y += e.y; v.z += e.z; v.w += e.w;
    __builtin_nontemporal_store(v, (f4*)out + t);      // streamed output
}

// ---------------------------------------------------------------------------
// Kernel 2: pxyz = predxyz * maskdiag   (307,200 elements -> 1200 blocks)
// ---------------------------------------------------------------------------
__global__ void __launch_bounds__(256) pxyz_kernel(const float* __restrict__ pred,
                                                   const float* __restrict__ md,
                                                   float* __restrict__ out)
{
    int t  = blockIdx.x * blockDim.x + threadIdx.x;   // 0 .. 307199
    int ij = t % LL;
    out[t] = pred[t] * md[ij];
}

// ---------------------------------------------------------------------------
// Kernel 3: cadistavg[j,k] = (1/L) * sum_i sqrt(||a_k - a_j||^2 + eps)
// with a_c = pred[c,i,col] * maskdiag[i,col].
// Squared distance via two chained V_WMMA_F32_16X16X4_F32 per (tile, i):
//   D  = A  x B   : A[j,:]=(x_j,y_j,z_j,1),  B[:,k]=(-2x_k,-2y_k,-2z_k,|a_k|^2)
//   D += A2 x B2  : A2[j,0]=|a_j|^2, B2[0,k]=1   (broadcasts |a_j|^2 per row)
// One wave per 16x16 tile; 20x20 tiles = 400 -> 100 blocks x 4 waves.
// Full waves, no divergence around WMMA -> EXEC all ones as required.
// Post-WMMA: v_med3 clamp + add + raw v_sqrt_f32 (3 VALU/element).
// ---------------------------------------------------------------------------
__global__ void __launch_bounds__(128) cad_kernel(const float* __restrict__ pred,
                                                  const float* __restrict__ md,
                                                  float* __restrict__ out)
{
    const int wave   = threadIdx.x >> 5;
    const int lane   = threadIdx.x & 31;
    const int lane15 = lane & 15;
    const bool hi    = (lane >= 16);

    const int tile = blockIdx.x * 4 + wave;       // 0..399
    const int tj   = tile / 20;
    const int tk   = tile - tj * 20;
    const int j0   = tj * 16;
    const int k0   = tk * 16;
    const int jc   = j0 + lane15;                 // this lane's j column
    const int kc   = k0 + lane15;                 // this lane's k column

    v8f acc = {};

    #pragma unroll 2
    for (int i = 0; i < L; ++i) {
        const float* row = pred + i * L;
        const float* mdr = md   + i * L;

        // masked coordinates (maskdiag zeros the i==col diagonal)
        float mdj = mdr[jc];
        float mdk = mdr[kc];
        float xj = row[jc]          * mdj;
        float yj = row[LL + jc]     * mdj;
        float zj = row[2 * LL + jc] * mdj;
        float xk = row[kc]          * mdk;
        float yk = row[LL + kc]     * mdk;
        float zk = row[2 * LL + kc] * mdk;

        float nj = xj * xj + yj * yj + zj * zj;
        float nk = xk * xk + yk * yk + zk * zk;

        // Pack per ISA 16x4 f32 A layout: lanes 0-15 -> (v0=K0, v1=K1),
        // lanes 16-31 -> (v0=K2, v1=K3). B mirrored by rows.
        v2f A, Bm, A2, B2;
        A.x  = hi ? zj : xj;                 // K2 = z_j       | K0 = x_j
        A.y  = hi ? 1.0f : yj;               // K3 = 1         | K1 = y_j
        Bm.x = -2.0f * (hi ? zk : xk);       // K2 = -2 z_k    | K0 = -2 x_k
        Bm.y = hi ? nk : (-2.0f * yk);       // K3 = |a_k|^2   | K1 = -2 y_k
        A2.x = hi ? 0.0f : nj;               // K2 = 0         | K0 = |a_j|^2
        A2.y = 0.0f;
        B2.x = hi ? 0.0f : 1.0f;             // K2 = 0         | K0 = 1
        B2.y = 0.0f;

        v8f d = {};
        d = __builtin_amdgcn_wmma_f32_16x16x4_f32(false, A,  false, Bm, (short)0, d, false, false);
        d = __builtin_amdgcn_wmma_f32_16x16x4_f32(false, A2, false, B2, (short)0, d, false, false);

        #pragma unroll
        for (int r = 0; r < 8; ++r) {
            // clamp tiny negative cancellation error with one v_med3_f32,
            // then raw hardware v_sqrt_f32 (~1 ulp, plenty here)
            float d2 = __builtin_amdgcn_fmed3f(d[r], 0.0f, 3.0e38f) + EPS;
            acc[r] += __builtin_amdgcn_sqrtf(d2);
        }
    }

    const float inv = 1.0f / (float)L;
    #pragma unroll
    for (int r = 0; r < 8; ++r) {
        // C/D layout: VGPR r holds row M = r (lanes 0-15) / r+8 (lanes 16-31)
        int jrow = j0 + r + (hi ? 8 : 0);
        out[jrow * L + k0 + lane15] = acc[r] * inv;
    }
}

// ---------------------------------------------------------------------------
extern "C" void kernel_launch(void* const* d_in, const int* in_sizes, int n_in,
                              void* d_out, int out_size, void* d_ws, size_t ws_size,
                              hipStream_t stream) {
    const int*           residx = (const int*)d_in[0];            // [1,320] int32
    const unsigned char* mask   = (const unsigned char*)d_in[1];  // [1,320] bool
    const float*         emb    = (const float*)d_in[2];          // [66,128] f32
    const float*         pf     = (const float*)d_in[3];          // [1,320,320,128] f32
    const float*         pred   = (const float*)d_in[4];          // [1,3,320,320] f32
    const float*         md     = (const float*)d_in[5];          // [1,1,320,320] f32
    float* out = (float*)d_out;

    // output layout (concatenated, return order): z | pxyz | cadistavg
    const long Z_OFF = 0;
    const long P_OFF = 13107200;   // 320*320*128
    const long C_OFF = 13414400;   // + 3*320*320

    z_kernel   <<<12800, 256, 0, stream>>>(residx, mask, emb, pf, out + Z_OFF);
    pxyz_kernel<<<1200,  256, 0, stream>>>(pred, md, out + P_OFF);
    cad_kernel <<<100,   128, 0, stream>>>(pred, md, out + C_OFF);
}